// TrajectoryCost_3487513444458
// MI455X (gfx1250) — compile-verified
//
#include <hip/hip_runtime.h>
#include <stdint.h>

#define GAS __attribute__((address_space(1)))
#define LAS __attribute__((address_space(3)))

typedef int v4i_t __attribute__((ext_vector_type(4)));

// Async global->LDS DMA path (CDNA5 ASYNCcnt path), probe-confirmed to lower
// to global_load_async_to_lds_b128 on this toolchain.
#if defined(__HIP_DEVICE_COMPILE__) && defined(__gfx1250__) && \
    __has_builtin(__builtin_amdgcn_global_load_async_to_lds_b128)
#define USE_ASYNC_LDS 1
#else
#define USE_ASYNC_LDS 0
#endif

__device__ __forceinline__ float safe_sqrtf(float x) {
  return x > 0.0f ? __builtin_sqrtf(x) : 0.0f;
}

#if USE_ASYNC_LDS
__device__ __forceinline__ void async_wait0() {
#if __has_builtin(__builtin_amdgcn_s_wait_asynccnt)
  __builtin_amdgcn_s_wait_asynccnt(0);
#else
  asm volatile("s_wait_asynccnt 0" ::: "memory");
#endif
}
#endif

// Per-item math. G[j][i] = Gv[j*3+i] (row j, col i), R[j][i] = R[j*3+i].
__device__ __forceinline__ void compute_item(
    const float* __restrict__ gpos, const float* __restrict__ grot,
    int h, const float R[9], float p0, float p1, float p2,
    float* __restrict__ out, long long BH, long long idx)
{
  float Gv[9];
#pragma unroll
  for (int i = 0; i < 9; ++i) Gv[i] = grot[(long long)h * 9 + i];
  const float q0 = gpos[(long long)h * 3 + 0];
  const float q1 = gpos[(long long)h * 3 + 1];
  const float q2 = gpos[(long long)h * 3 + 2];

  // t[i] = -(G^T g)[i] + (R^T p)[i]
  const float t0 = -(Gv[0]*q0 + Gv[3]*q1 + Gv[6]*q2) + (R[0]*p0 + R[3]*p1 + R[6]*p2);
  const float t1 = -(Gv[1]*q0 + Gv[4]*q1 + Gv[7]*q2) + (R[1]*p0 + R[4]*p1 + R[7]*p2);
  const float t2 = -(Gv[2]*q0 + Gv[5]*q1 + Gv[8]*q2) + (R[2]*p0 + R[5]*p1 + R[8]*p2);
  const float pos_err   = t0*t0 + t1*t1 + t2*t2;
  const float goal_dist = safe_sqrtf(pos_err);

  // M[i][j] = sum_k G[k][i]*R[k][j];  d = I - M;  rn[i] = ||d[i,:]||
  float rn[3];
#pragma unroll
  for (int i = 0; i < 3; ++i) {
    float acc = 0.0f;
#pragma unroll
    for (int j = 0; j < 3; ++j) {
      const float m = Gv[i]*R[j] + Gv[3+i]*R[3+j] + Gv[6+i]*R[6+j];
      const float d = ((i == j) ? 1.0f : 0.0f) - m;
      acc += d * d;
    }
    rn[i] = safe_sqrtf(acc);
  }

  const float rot_err_norm = safe_sqrtf(rn[0]*rn[0] + rn[1]*rn[1] + rn[2]*rn[2]);
  const float srn = rn[0] + rn[1] + rn[2];
  float rot_err = srn * srn;                         // rot_weight == 1
  rot_err = (goal_dist <= 100.0f) ? rot_err : 0.0f;  // hinge
  const float cost = safe_sqrtf(rot_err) + safe_sqrtf(pos_err);  // WEIGHT == (1,1)

  out[idx]          = cost;
  out[BH + idx]     = rot_err_norm;
  out[2 * BH + idx] = goal_dist;
}

// Stage a full 256-item tile (rot: 9216 B, pos: 3072 B) into LDS.
// Async path: straight-line issue from one VGPR address pair; the 24-bit
// instruction offset is added to BOTH global and LDS addresses (ISA §15.18.3),
// so offsets 0/4096/8192 cover the rot tile with no loop or extra addr math.
__device__ __forceinline__ void stage_tile(
    const float* __restrict__ pos_b, const float* __restrict__ rot_b,
    long long base, int tid, float* s_rot, float* s_pos)
{
#if USE_ASYNC_LDS
  GAS v4i_t* grot_t = (GAS v4i_t*)(rot_b + base * 9) + tid;
  GAS v4i_t* gpos_t = (GAS v4i_t*)(pos_b + base * 3) + tid;
  LAS v4i_t* lrot   = (LAS v4i_t*)s_rot + tid;
  LAS v4i_t* lpos   = (LAS v4i_t*)s_pos + tid;
  __builtin_amdgcn_global_load_async_to_lds_b128(grot_t, lrot, 0,    0);
  __builtin_amdgcn_global_load_async_to_lds_b128(grot_t, lrot, 4096, 0);
  if (tid < 64)
    __builtin_amdgcn_global_load_async_to_lds_b128(grot_t, lrot, 8192, 0);
  if (tid < 192)
    __builtin_amdgcn_global_load_async_to_lds_b128(gpos_t, lpos, 0, 0);
  async_wait0();       // drain this wave's ASYNCcnt
  __syncthreads();     // make all waves' tiles visible
#else
  const float4* gr4 = (const float4*)(rot_b + base * 9);
  float4* sr4 = (float4*)s_rot;
  sr4[tid]       = gr4[tid];
  sr4[tid + 256] = gr4[tid + 256];
  if (tid < 64) sr4[tid + 512] = gr4[tid + 512];
  if (tid < 192) ((float4*)s_pos)[tid] = ((const float4*)(pos_b + base * 3))[tid];
  __syncthreads();
#endif
}

// Fast path: H % 256 == 0. grid = (H/256, B); h needs no modulo.
__global__ __launch_bounds__(256) void traj_cost_kernel_2d(
    const float* __restrict__ pos_b, const float* __restrict__ rot_b,
    const float* __restrict__ gpos, const float* __restrict__ grot,
    float* __restrict__ out, int H, long long BH)
{
  __shared__ __align__(16) float s_rot[256 * 9];
  __shared__ __align__(16) float s_pos[256 * 3];

  const int tid = threadIdx.x;
  const int h = blockIdx.x * 256 + tid;
  const long long base = (long long)blockIdx.y * H + (long long)blockIdx.x * 256;
  const long long idx = base + tid;

  stage_tile(pos_b, rot_b, base, tid, s_rot, s_pos);

  float R[9];
#pragma unroll
  for (int i = 0; i < 9; ++i) R[i] = s_rot[tid * 9 + i];  // stride 9: conflict-free
  const float p0 = s_pos[tid * 3 + 0];
  const float p1 = s_pos[tid * 3 + 1];
  const float p2 = s_pos[tid * 3 + 2];

  compute_item(gpos, grot, h, R, p0, p1, p2, out, BH, idx);
}

// Generic fallback: 1D grid, runtime modulo for h, handles partial tail tile.
__global__ __launch_bounds__(256) void traj_cost_kernel_1d(
    const float* __restrict__ pos_b, const float* __restrict__ rot_b,
    const float* __restrict__ gpos, const float* __restrict__ grot,
    float* __restrict__ out, int H, long long BH)
{
  __shared__ __align__(16) float s_rot[256 * 9];
  __shared__ __align__(16) float s_pos[256 * 3];

  const int tid = threadIdx.x;
  const long long base = (long long)blockIdx.x * 256;
  const bool full = (base + 256) <= BH;

  if (full) stage_tile(pos_b, rot_b, base, tid, s_rot, s_pos);

  const long long idx = base + tid;
  if (idx < BH) {
    const int h = (int)(idx % H);
    float R[9], p0, p1, p2;
    if (full) {
#pragma unroll
      for (int i = 0; i < 9; ++i) R[i] = s_rot[tid * 9 + i];
      p0 = s_pos[tid * 3 + 0];
      p1 = s_pos[tid * 3 + 1];
      p2 = s_pos[tid * 3 + 2];
    } else {
      const float* rp = rot_b + idx * 9;
#pragma unroll
      for (int i = 0; i < 9; ++i) R[i] = rp[i];
      const float* pp = pos_b + idx * 3;
      p0 = pp[0]; p1 = pp[1]; p2 = pp[2];
    }
    compute_item(gpos, grot, h, R, p0, p1, p2, out, BH, idx);
  }
}

extern "C" void kernel_launch(void* const* d_in, const int* in_sizes, int n_in,
                              void* d_out, int out_size, void* d_ws, size_t ws_size,
                              hipStream_t stream) {
  (void)n_in; (void)out_size; (void)d_ws; (void)ws_size;
  const float* pos_b = (const float*)d_in[0];  // [B,H,3]
  const float* rot_b = (const float*)d_in[1];  // [B,H,3,3]
  const float* gpos  = (const float*)d_in[2];  // [H,3]
  const float* grot  = (const float*)d_in[3];  // [H,3,3]
  float* out = (float*)d_out;

  const int H = in_sizes[2] / 3;
  const long long BH = (long long)in_sizes[0] / 3;

  if (H > 0 && (H % 256) == 0) {
    const long long B = BH / H;
    dim3 grid((unsigned)(H / 256), (unsigned)B);
    traj_cost_kernel_2d<<<grid, dim3(256), 0, stream>>>(
        pos_b, rot_b, gpos, grot, out, H, BH);
  } else {
    const long long nBlocks = (BH + 255) / 256;
    traj_cost_kernel_1d<<<dim3((unsigned)nBlocks), dim3(256), 0, stream>>>(
        pos_b, rot_b, gpos, grot, out, H, BH);
  }
}